// ProteinGNN_77567109366206
// MI455X (gfx1250) — compile-verified
//
#include <hip/hip_runtime.h>

#define NN    50000
#define DIN   32
#define HID   64
#define HEADS 4
#define DOUT  32
#define FEAT  (HEADS * HID)   // 256
#define NEG_SLOPE 0.2f

typedef __attribute__((ext_vector_type(2))) float v2f;
typedef __attribute__((ext_vector_type(8))) float v8f;

__device__ __forceinline__ v8f wmma4(v2f a, v2f b, v8f c) {
  // D = A(16x4,f32) * B(4x16,f32) + C(16x16,f32)
  return __builtin_amdgcn_wmma_f32_16x16x4_f32(false, a, false, b, (short)0, c, false, false);
}

__device__ __forceinline__ v8f zero8() {
  v8f z;
#pragma unroll
  for (int i = 0; i < 8; i++) z[i] = 0.0f;
  return z;
}

// monotonic float<->uint mapping for atomicMax-based float segment-max
__device__ __forceinline__ unsigned fenc(float f) {
  unsigned u = __float_as_uint(f);
  return (u & 0x80000000u) ? ~u : (u | 0x80000000u);
}
__device__ __forceinline__ float fdec(unsigned u) {
  return (u & 0x80000000u) ? __uint_as_float(u & 0x7fffffffu) : __uint_as_float(~u);
}

// ---------------- init ----------------
__global__ void k_init(float* out2, float* hacc, float* denom, unsigned* menc, float* deg) {
  int i = blockIdx.x * blockDim.x + threadIdx.x;
  if (i < NN * FEAT) out2[i] = 0.0f;
  if (i < NN * HID) hacc[i] = 0.0f;
  if (i < NN * HEADS) { denom[i] = 0.0f; menc[i] = 0u; }
  if (i < NN) deg[i] = 1.0f;  // self-loop contribution
}

// ---------------- GEMM: hg = x(N,32) @ W_gcn(32,64) ----------------
__global__ void k_gemm_gcn(const float* __restrict__ x, const float* __restrict__ W,
                           float* __restrict__ hg) {
  int wave = (blockIdx.x * blockDim.x + threadIdx.x) >> 5;
  int lane = threadIdx.x & 31;
  int row0 = wave * 16;
  if (row0 >= NN) return;
  int half = lane >> 4, lm = lane & 15;
  const float* xr = x + (size_t)(row0 + lm) * DIN;
  v2f afr[8];
#pragma unroll
  for (int ks = 0; ks < 8; ks++) {
    int k0 = ks * 4 + 2 * half;
    afr[ks] = (v2f){xr[k0], xr[k0 + 1]};
  }
#pragma unroll
  for (int nt = 0; nt < 4; nt++) {
    v8f c = zero8();
#pragma unroll
    for (int ks = 0; ks < 8; ks++) {
      int k0 = ks * 4 + 2 * half;
      v2f b = (v2f){W[k0 * HID + nt * 16 + lm], W[(k0 + 1) * HID + nt * 16 + lm]};
      c = wmma4(afr[ks], b, c);
    }
#pragma unroll
    for (int r = 0; r < 8; r++)
      hg[(size_t)(row0 + r + half * 8) * HID + nt * 16 + lm] = c[r];
  }
}

// ---------------- degree ----------------
__global__ void k_deg(const int* __restrict__ ei, int E, float* deg) {
  int e = blockIdx.x * blockDim.x + threadIdx.x;
  if (e >= E) return;
  atomicAdd(&deg[ei[E + e]], 1.0f);
}
__global__ void k_dinv(float* deg) {
  int i = blockIdx.x * blockDim.x + threadIdx.x;
  if (i >= NN) return;
  deg[i] = rsqrtf(deg[i]);  // deg >= 1 always (self-loop)
}

// ---------------- GCN aggregation: wave per edge ----------------
__global__ void k_gcn_agg(const int* __restrict__ ei, int E, const float* __restrict__ dinv,
                          const float* __restrict__ hg, float* __restrict__ hacc) {
  int wid = (blockIdx.x * blockDim.x + threadIdx.x) >> 5;
  int lane = threadIdx.x & 31;
  if (wid >= E + NN) return;
  int s, d;
  if (wid < E) { s = ei[wid]; d = ei[E + wid]; } else { s = d = wid - E; }
  float norm = dinv[s] * dinv[d];
  const float* hs = hg + (size_t)s * HID;
  float* hd = hacc + (size_t)d * HID;
  atomicAdd(&hd[lane], hs[lane] * norm);
  atomicAdd(&hd[lane + 32], hs[lane + 32] * norm);
}

// ---------------- bias + relu -> h (into hg buffer) ----------------
__global__ void k_biasrelu(const float* __restrict__ hacc, const float* __restrict__ b,
                           float* __restrict__ h) {
  int i = blockIdx.x * blockDim.x + threadIdx.x;
  if (i >= NN * HID) return;
  h[i] = fmaxf(hacc[i] + b[i & (HID - 1)], 0.0f);
}

// ---------------- GEMM: hgat = h(N,64) @ W_gat(64,256) ----------------
__global__ void k_gemm_gat(const float* __restrict__ h, const float* __restrict__ W,
                           float* __restrict__ hgat) {
  int wave = (blockIdx.x * blockDim.x + threadIdx.x) >> 5;
  int lane = threadIdx.x & 31;
  int row0 = wave * 16;
  if (row0 >= NN) return;
  int half = lane >> 4, lm = lane & 15;
  const float* hr = h + (size_t)(row0 + lm) * HID;
  v2f afr[16];
#pragma unroll
  for (int ks = 0; ks < 16; ks++) {
    int k0 = ks * 4 + 2 * half;
    afr[ks] = (v2f){hr[k0], hr[k0 + 1]};
  }
  for (int nt = 0; nt < 16; nt++) {
    v8f c = zero8();
#pragma unroll
    for (int ks = 0; ks < 16; ks++) {
      int k0 = ks * 4 + 2 * half;
      v2f b = (v2f){W[k0 * FEAT + nt * 16 + lm], W[(k0 + 1) * FEAT + nt * 16 + lm]};
      c = wmma4(afr[ks], b, c);
    }
#pragma unroll
    for (int r = 0; r < 8; r++)
      hgat[(size_t)(row0 + r + half * 8) * FEAT + nt * 16 + lm] = c[r];
  }
}

// ---------------- attention logits per (node, head) ----------------
__global__ void k_att(const float* __restrict__ hgat, const float* __restrict__ att_s,
                      const float* __restrict__ att_d, float* __restrict__ as_,
                      float* __restrict__ ad_) {
  int i = blockIdx.x * blockDim.x + threadIdx.x;  // node*HEADS + h
  if (i >= NN * HEADS) return;
  int hh = i & (HEADS - 1);
  const float* hr = hgat + (size_t)(i >> 2) * FEAT + hh * HID;
  const float* vs = att_s + hh * HID;
  const float* vd = att_d + hh * HID;
  float ss = 0.0f, sd = 0.0f;
#pragma unroll 8
  for (int c = 0; c < HID; c++) {
    float v = hr[c];
    ss += v * vs[c];
    sd += v * vd[c];
  }
  as_[i] = ss;
  ad_[i] = sd;
}

__device__ __forceinline__ void edge_sd(const int* ei, int E, int e, int& s, int& d) {
  if (e < E) { s = ei[e]; d = ei[E + e]; } else { s = d = e - E; }
}

// ---------------- segment-max of leaky-relu logits ----------------
__global__ void k_attmax(const int* __restrict__ ei, int E, const float* __restrict__ as_,
                         const float* __restrict__ ad_, unsigned* __restrict__ menc) {
  int e = blockIdx.x * blockDim.x + threadIdx.x;
  if (e >= E + NN) return;
  int s, d; edge_sd(ei, E, e, s, d);
#pragma unroll
  for (int h = 0; h < HEADS; h++) {
    float v = as_[s * HEADS + h] + ad_[d * HEADS + h];
    v = v > 0.0f ? v : NEG_SLOPE * v;
    atomicMax(&menc[d * HEADS + h], fenc(v));
  }
}

// ---------------- segment-sum of exp(e - max) ----------------
__global__ void k_attsum(const int* __restrict__ ei, int E, const float* __restrict__ as_,
                         const float* __restrict__ ad_, const unsigned* __restrict__ menc,
                         float* __restrict__ denom) {
  int e = blockIdx.x * blockDim.x + threadIdx.x;
  if (e >= E + NN) return;
  int s, d; edge_sd(ei, E, e, s, d);
#pragma unroll
  for (int h = 0; h < HEADS; h++) {
    float v = as_[s * HEADS + h] + ad_[d * HEADS + h];
    v = v > 0.0f ? v : NEG_SLOPE * v;
    float w = __expf(v - fdec(menc[d * HEADS + h]));
    atomicAdd(&denom[d * HEADS + h], w);
  }
}

// ---------------- GAT aggregation: wave per edge, 256 channels ----------------
__global__ void k_gat_agg(const int* __restrict__ ei, int E, const float* __restrict__ as_,
                          const float* __restrict__ ad_, const unsigned* __restrict__ menc,
                          const float* __restrict__ denom, const float* __restrict__ hgat,
                          float* __restrict__ out2) {
  int wid = (blockIdx.x * blockDim.x + threadIdx.x) >> 5;
  int lane = threadIdx.x & 31;
  if (wid >= E + NN) return;
  int s, d; edge_sd(ei, E, wid, s, d);
  float alpha[HEADS];
#pragma unroll
  for (int h = 0; h < HEADS; h++) {
    float v = as_[s * HEADS + h] + ad_[d * HEADS + h];
    v = v > 0.0f ? v : NEG_SLOPE * v;
    float w = __expf(v - fdec(menc[d * HEADS + h]));
    alpha[h] = w / denom[d * HEADS + h];
  }
  const float* hs = hgat + (size_t)s * FEAT;
  float* od = out2 + (size_t)d * FEAT;
#pragma unroll
  for (int j = 0; j < 8; j++) {
    int c = lane + 32 * j;            // coalesced 128B per j
    atomicAdd(&od[c], hs[c] * alpha[j >> 1]);  // head = c>>6 = j>>1
  }
}

// ---------------- final: out = relu(out2 + b_gat) @ W_fc + b_fc ----------------
__global__ void k_fc(const float* __restrict__ out2, const float* __restrict__ bg,
                     const float* __restrict__ W, const float* __restrict__ bf,
                     float* __restrict__ out) {
  int wave = (blockIdx.x * blockDim.x + threadIdx.x) >> 5;
  int lane = threadIdx.x & 31;
  int row0 = wave * 16;
  if (row0 >= NN) return;
  int half = lane >> 4, lm = lane & 15;
  const float* ar = out2 + (size_t)(row0 + lm) * FEAT;
  v8f c0 = zero8(), c1 = zero8();
#pragma unroll 4
  for (int ks = 0; ks < 64; ks++) {
    int k0 = ks * 4 + 2 * half;
    v2f a = (v2f){fmaxf(ar[k0] + bg[k0], 0.0f), fmaxf(ar[k0 + 1] + bg[k0 + 1], 0.0f)};
    v2f b0 = (v2f){W[k0 * DOUT + lm], W[(k0 + 1) * DOUT + lm]};
    v2f b1 = (v2f){W[k0 * DOUT + 16 + lm], W[(k0 + 1) * DOUT + 16 + lm]};
    c0 = wmma4(a, b0, c0);
    c1 = wmma4(a, b1, c1);
  }
#pragma unroll
  for (int r = 0; r < 8; r++) {
    int row = row0 + r + half * 8;
    out[(size_t)row * DOUT + lm] = c0[r] + bf[lm];
    out[(size_t)row * DOUT + 16 + lm] = c1[r] + bf[16 + lm];
  }
}

extern "C" void kernel_launch(void* const* d_in, const int* in_sizes, int n_in,
                              void* d_out, int out_size, void* d_ws, size_t ws_size,
                              hipStream_t stream) {
  const float* x     = (const float*)d_in[0];
  const int*   ei    = (const int*)d_in[1];
  const float* Wg    = (const float*)d_in[2];
  const float* bg    = (const float*)d_in[3];
  const float* Wa    = (const float*)d_in[4];
  const float* att_s = (const float*)d_in[5];
  const float* att_d = (const float*)d_in[6];
  const float* ba    = (const float*)d_in[7];
  const float* Wf    = (const float*)d_in[8];
  const float* bf    = (const float*)d_in[9];
  float* out = (float*)d_out;
  int E = in_sizes[1] / 2;

  float* ws = (float*)d_ws;
  float*    hg    = ws;                       // N*64, then reused for relu'd GCN out
  float*    hacc  = hg + (size_t)NN * HID;    // N*64
  float*    deg   = hacc + (size_t)NN * HID;  // N (becomes dinv in place)
  float*    as_   = deg + NN;                 // N*4
  float*    ad_   = as_ + NN * HEADS;         // N*4
  unsigned* menc  = (unsigned*)(ad_ + NN * HEADS);  // N*4
  float*    denom = (float*)(menc + NN * HEADS);    // N*4
  float*    hgat  = denom + NN * HEADS;       // N*256
  float*    out2  = hgat + (size_t)NN * FEAT; // N*256

  const int TPB = 256;
  int tiles = NN / 16;          // 3125 exact
  int EW = E + NN;              // edges incl. self-loops

  k_init<<<(NN * FEAT + TPB - 1) / TPB, TPB, 0, stream>>>(out2, hacc, denom, menc, deg);
  k_gemm_gcn<<<(tiles + 7) / 8, TPB, 0, stream>>>(x, Wg, hg);
  k_deg<<<(E + TPB - 1) / TPB, TPB, 0, stream>>>(ei, E, deg);
  k_dinv<<<(NN + TPB - 1) / TPB, TPB, 0, stream>>>(deg);
  k_gcn_agg<<<(EW + 7) / 8, TPB, 0, stream>>>(ei, E, deg, hg, hacc);
  k_biasrelu<<<(NN * HID + TPB - 1) / TPB, TPB, 0, stream>>>(hacc, bg, hg);
  k_gemm_gat<<<(tiles + 7) / 8, TPB, 0, stream>>>(hg, Wa, hgat);
  k_att<<<(NN * HEADS + TPB - 1) / TPB, TPB, 0, stream>>>(hgat, att_s, att_d, as_, ad_);
  k_attmax<<<(EW + TPB - 1) / TPB, TPB, 0, stream>>>(ei, E, as_, ad_, menc);
  k_attsum<<<(EW + TPB - 1) / TPB, TPB, 0, stream>>>(ei, E, as_, ad_, menc, denom);
  k_gat_agg<<<(EW + 7) / 8, TPB, 0, stream>>>(ei, E, as_, ad_, menc, denom, hgat, out2);
  k_fc<<<(tiles + 7) / 8, TPB, 0, stream>>>(out2, ba, Wf, bf, out);
}